// Copy_SOCNet_67164698575357
// MI455X (gfx1250) — compile-verified
//
#include <hip/hip_runtime.h>
#include <math.h>

#define B_DIM   1024
#define T_DIM   8192
#define TILE    1024
#define NTILES  (T_DIM / TILE)        // 8
#define THREADS 256
#define NWAVES  (THREADS / 32)        // 8
#define EPT     (TILE / THREADS)      // 4 elements per thread
#define TILE_FLOATS (TILE * 4 + 4)    // 4100 floats: tile + one extra elem for ts[t+1]

typedef unsigned int v4u __attribute__((ext_vector_type(4)));
typedef int          v8i __attribute__((ext_vector_type(8)));
typedef int          v4i __attribute__((ext_vector_type(4)));

#if defined(__has_builtin)
#if __has_builtin(__builtin_amdgcn_tensor_load_to_lds)
#define HAS_TDM 1
#endif
#endif

// Precise softplus (used once per block for SOC_init).
__device__ __forceinline__ float softplus_precise(float x) {
    return fmaxf(x, 0.0f) + log1pf(expf(-fabsf(x)));
}

// Fast softplus for the hot loop: v_exp_f32 / v_log_f32.
// log argument is in [1,2] -> hw log accuracy is fine; exp arg <= 0.
__device__ __forceinline__ float softplus_fast(float x) {
    return fmaxf(x, 0.0f) + __logf(1.0f + __expf(-fabsf(x)));
}

#ifdef HAS_TDM
// Issue a TDM 1-D copy of tile_floats f32 elements from gptr into LDS byte
// offset lds_off. Reads past valid_floats return 0 (tensor_dim0 OOB rule).
// D# layout per CDNA5 ISA sec. 8.3/8.4.
__device__ __forceinline__ void tdm_load_tile(const float* gptr, unsigned lds_off,
                                              unsigned valid_floats, unsigned tile_floats) {
    unsigned long long ga = (unsigned long long)(uintptr_t)gptr;
    v4u g0;
    g0[0] = 1u;                                            // count=1, user descriptor
    g0[1] = lds_off;                                       // lds_addr
    g0[2] = (unsigned)(ga & 0xFFFFFFFFull);                // global_addr[31:0]
    g0[3] = (unsigned)((ga >> 32) & 0x1FFFFFFull)          // global_addr[56:32]
          | 0x80000000u;                                   // type=2 ("image")
    v8i g1;
    g1[0] = 0x20000;                                       // data_size=2 (4 bytes), mask=0
    g1[1] = (int)((valid_floats & 0xFFFFu) << 16);         // tensor_dim0[15:0]
    g1[2] = (int)((valid_floats >> 16) | (1u << 16));      // tensor_dim0[31:16] | tensor_dim1=1
    g1[3] = (int)(tile_floats << 16);                      // tile_dim0
    g1[4] = 1;                                             // tile_dim1=1, tile_dim2=0
    g1[5] = 0; g1[6] = 0; g1[7] = 0;                       // strides unused (1 line)
    v4i z4 = {0, 0, 0, 0};
#if __clang_major__ >= 23
    v8i z8 = {0, 0, 0, 0, 0, 0, 0, 0};
    __builtin_amdgcn_tensor_load_to_lds(g0, g1, z4, z4, z8, 0);
#else
    __builtin_amdgcn_tensor_load_to_lds(g0, g1, z4, z4, 0);
#endif
}
#endif

__global__ __launch_bounds__(THREADS)
void soc_scan_kernel(const float* __restrict__ X,  const float* __restrict__ SC,
                     const float* __restrict__ W1i, const float* __restrict__ b1i,
                     const float* __restrict__ W2i, const float* __restrict__ b2i,
                     const float* __restrict__ W1e, const float* __restrict__ b1e,
                     const float* __restrict__ W2e, const float* __restrict__ b2e,
                     float* __restrict__ out) {
    __shared__ __align__(16) float lds[2][TILE_FLOATS];
    __shared__ __align__(16) float wsum[NWAVES];

    const int b    = blockIdx.x;
    const int tid  = threadIdx.x;
    const int lane = tid & 31;
    const int wid  = tid >> 5;
    const size_t rowX = (size_t)b * T_DIM * 4;

    // --- per-row uniforms (broadcast loads, once per block) ---
    const float w1e0 = W1e[0], w1e1 = W1e[1], b1ev = b1e[0];
    const float w2ev = W2e[0], b2ev = b2e[0];
    const float Q    = SC[b * 4 + 0];
    const float eta0 = SC[b * 4 + 1];
    const float R    = SC[b * 4 + 2];
    const float S3   = SC[b * 4 + 3];

    // SOC_init from first timestep's features + R
    const float I0 = X[rowX + 1], T0 = X[rowX + 2], U0 = X[rowX + 3];
    const float z0 = I0 * W1i[0] + T0 * W1i[1] + U0 * W1i[2] + R * W1i[3] + b1i[0];
    const float soc_net  = softplus_precise(z0) * W2i[0] + b2i[0];
    const float soc_init = S3 * (1.0f + soc_net);
    const float kf = eta0 / (3600.0f * Q);              // eta0 / (3600 Q)

    const unsigned total_floats = (unsigned)B_DIM * (unsigned)T_DIM * 4u;

#ifdef HAS_TDM
    const unsigned lds_off0 = (unsigned)(uintptr_t)&lds[0][0];
    const unsigned lds_off1 = (unsigned)(uintptr_t)&lds[1][0];
    if (wid == 0) {
        unsigned sf = (unsigned)b * (unsigned)T_DIM * 4u;
        tdm_load_tile(X + sf, lds_off0, total_floats - sf, TILE_FLOATS);
    }
#endif

    float carry = 0.0f;                                  // uniform across all threads
    for (int it = 0; it < NTILES; ++it) {
        const int base = it * TILE;
#ifdef HAS_TDM
        const float* buf = lds[it & 1];
        if (wid == 0) {
            if (it + 1 < NTILES) {
                unsigned sf = ((unsigned)b * (unsigned)T_DIM + (unsigned)(it + 1) * TILE) * 4u;
                tdm_load_tile(X + sf, (it & 1) ? lds_off0 : lds_off1,
                              total_floats - sf, TILE_FLOATS);
                __builtin_amdgcn_s_wait_tensorcnt(1);    // current tile complete
            } else {
                __builtin_amdgcn_s_wait_tensorcnt(0);
            }
        }
        __syncthreads();                                 // (A) tile data visible to all waves
#else
        const float* buf = lds[0];
        __syncthreads();                                 // protect buffer reuse
        {
            unsigned sf = ((unsigned)b * (unsigned)T_DIM + (unsigned)base) * 4u;
            const unsigned nvec = TILE_FLOATS / 4;       // 1025 float4 chunks
            for (unsigned i = tid; i < nvec; i += THREADS) {
                float4 v = {0.f, 0.f, 0.f, 0.f};
                if (sf + i * 4u + 3u < total_floats)
                    v = ((const float4*)(X + sf))[i];
                ((float4*)&lds[0][0])[i] = v;
            }
        }
        __syncthreads();
#endif

        // --- per-thread: EPT elements, serial inclusive scan ---
        float g[EPT];
        float s = 0.0f;
#pragma unroll
        for (int j = 0; j < EPT; ++j) {
            const int e = tid * EPT + j;                 // local element in tile
            const float4 v = ((const float4*)buf)[e];    // {ts, I, Temp, U}
            const float ts_next = buf[(e + 1) * 4];
            const float h  = softplus_fast(v.y * w1e0 + v.z * w1e1 + b1ev);
            const float de = h * w2ev + b2ev;
            const float f  = kf * (1.0f + de) * v.y;
            const int te = base + e;
            const float gv = (te < T_DIM - 1) ? (ts_next - v.x) * f : 0.0f;
            s += gv;
            g[j] = s;                                    // inclusive within thread
        }

        // --- wave32 inclusive scan of per-thread totals ---
        float ws = s;
#pragma unroll
        for (int d = 1; d < 32; d <<= 1) {
            float n = __shfl_up(ws, (unsigned)d, 32);
            if (lane >= d) ws += n;
        }
        if (lane == 31) wsum[wid] = ws;
        __syncthreads();                                 // (B) wave sums visible

        // two ds_load_b128 broadcast reads instead of 8 scalar LDS reads
        const float4 wa = ((const float4*)wsum)[0];
        const float4 wb = ((const float4*)wsum)[1];
        const float wv[NWAVES] = {wa.x, wa.y, wa.z, wa.w, wb.x, wb.y, wb.z, wb.w};
        float wave_off = 0.0f, tile_total = 0.0f;
#pragma unroll
        for (int w = 0; w < NWAVES; ++w) {
            if (w < wid) wave_off += wv[w];
            tile_total += wv[w];
        }

        const float thread_excl = carry + wave_off + (ws - s);
        float4 o;
        o.x = soc_init + thread_excl;                    // exclusive prefix
        o.y = soc_init + thread_excl + g[0];
        o.z = soc_init + thread_excl + g[1];
        o.w = soc_init + thread_excl + g[2];
        ((float4*)(out + (size_t)b * T_DIM + base))[tid] = o;

        carry += tile_total;                             // uniform update
    }
}

extern "C" void kernel_launch(void* const* d_in, const int* in_sizes, int n_in,
                              void* d_out, int out_size, void* d_ws, size_t ws_size,
                              hipStream_t stream) {
    (void)in_sizes; (void)n_in; (void)d_ws; (void)ws_size; (void)out_size;
    const float* X   = (const float*)d_in[0];
    const float* SC  = (const float*)d_in[1];
    const float* W1i = (const float*)d_in[2];
    const float* b1i = (const float*)d_in[3];
    const float* W2i = (const float*)d_in[4];
    const float* b2i = (const float*)d_in[5];
    const float* W1e = (const float*)d_in[6];
    const float* b1e = (const float*)d_in[7];
    const float* W2e = (const float*)d_in[8];
    const float* b2e = (const float*)d_in[9];
    soc_scan_kernel<<<dim3(B_DIM), dim3(THREADS), 0, stream>>>(
        X, SC, W1i, b1i, W2i, b2i, W1e, b1e, W2e, b2e, (float*)d_out);
}